// StandGCN1_15839839387789
// MI455X (gfx1250) — compile-verified
//
#include <hip/hip_runtime.h>

typedef float v2f __attribute__((ext_vector_type(2)));
typedef float v8f __attribute__((ext_vector_type(8)));

#define N_NODES 100000
#define N_EDGES 3200000
#define NFEAT   256
#define NCLASS  64
#define MTILES  (N_NODES / 16)   // 6250 exact
#define WAVES_PER_BLOCK 8

// ---------------- degree / norm ----------------

__global__ void k_init_deg(float* __restrict__ deg) {
    int i = blockIdx.x * blockDim.x + threadIdx.x;
    if (i < N_NODES) deg[i] = 1.0f;           // self-loop contributes 1
}

__global__ void k_count(const int* __restrict__ dst, float* __restrict__ deg) {
    int e = blockIdx.x * blockDim.x + threadIdx.x;
    if (e < N_EDGES) atomicAdd(&deg[dst[e]], 1.0f);
}

__global__ void k_dinv(float* __restrict__ deg) {
    int i = blockIdx.x * blockDim.x + threadIdx.x;
    if (i < N_NODES) deg[i] = __frsqrt_rn(deg[i]);   // deg >= 1 always
}

// ---------------- WMMA fp32 GEMM: xw = x @ W ----------------
// One wave computes a 16x64 output tile with 4x v_wmma_f32_16x16x4_f32 per K-step.

__global__ __launch_bounds__(256) void k_gemm_wmma(const float* __restrict__ x,
                                                   const float* __restrict__ W,
                                                   float* __restrict__ xw) {
    // Stage W transposed into LDS: Wt[c*256 + k] = W[k*64 + c]  (64 KB)
    __shared__ float Wt[NCLASS * NFEAT];
    for (int i = threadIdx.x; i < NCLASS * NFEAT; i += 256) {
        int k = i >> 6;
        int c = i & 63;
        Wt[c * NFEAT + k] = W[i];
    }
    __syncthreads();

    const int lane  = threadIdx.x & 31;
    const int wave  = threadIdx.x >> 5;
    const int mtile = blockIdx.x * WAVES_PER_BLOCK + wave;
    if (mtile >= MTILES) return;              // wave-uniform: EXEC stays all-1s

    const int half = lane >> 4;               // 0 or 1
    const int l15  = lane & 15;
    const float* xrow = x + (size_t)(mtile * 16 + l15) * NFEAT;

    v8f acc0 = {}, acc1 = {}, acc2 = {}, acc3 = {};

    for (int k = 0; k < NFEAT; k += 4) {
        const int ka = k + 2 * half;          // A: lane holds K = ka, ka+1
        v2f a  = *(const v2f*)(xrow + ka);
        v2f b0 = *(const v2f*)(&Wt[(0 * 16 + l15) * NFEAT + ka]);
        v2f b1 = *(const v2f*)(&Wt[(1 * 16 + l15) * NFEAT + ka]);
        v2f b2 = *(const v2f*)(&Wt[(2 * 16 + l15) * NFEAT + ka]);
        v2f b3 = *(const v2f*)(&Wt[(3 * 16 + l15) * NFEAT + ka]);
        acc0 = __builtin_amdgcn_wmma_f32_16x16x4_f32(false, a, false, b0, (short)0, acc0, false, false);
        acc1 = __builtin_amdgcn_wmma_f32_16x16x4_f32(false, a, false, b1, (short)0, acc1, false, false);
        acc2 = __builtin_amdgcn_wmma_f32_16x16x4_f32(false, a, false, b2, (short)0, acc2, false, false);
        acc3 = __builtin_amdgcn_wmma_f32_16x16x4_f32(false, a, false, b3, (short)0, acc3, false, false);
    }

    // C/D layout: VGPR r -> M = r + 8*half, N = l15 (+16*ntile)
    float* orow = xw + (size_t)mtile * 16 * NCLASS;
#pragma unroll
    for (int r = 0; r < 8; ++r) {
        int m = r + 8 * half;
        orow[m * NCLASS + 0 * 16 + l15] = acc0[r];
        orow[m * NCLASS + 1 * 16 + l15] = acc1[r];
        orow[m * NCLASS + 2 * 16 + l15] = acc2[r];
        orow[m * NCLASS + 3 * 16 + l15] = acc3[r];
    }
}

// ---------------- out init: bias + self-loop term ----------------

__global__ void k_out_init(const float* __restrict__ xw, const float* __restrict__ dinv,
                           const float* __restrict__ bias, float* __restrict__ out) {
    int i = blockIdx.x * blockDim.x + threadIdx.x;
    if (i < N_NODES * NCLASS) {
        int n = i >> 6;
        int c = i & 63;
        float di = dinv[n];
        out[i] = bias[c] + xw[i] * di * di;
    }
}

// ---------------- edge scatter-add (L2-resident atomics) ----------------

__global__ void k_scatter(const int* __restrict__ src, const int* __restrict__ dst,
                          const float* __restrict__ xw, const float* __restrict__ dinv,
                          float* __restrict__ out) {
    long long t = (long long)blockIdx.x * blockDim.x + threadIdx.x;
    if (t >= (long long)N_EDGES * NCLASS) return;
    int e = (int)(t >> 6);
    int c = (int)(t & 63);
    int s = src[e];
    int d = dst[e];
    float norm = dinv[s] * dinv[d];
    atomicAdd(&out[(size_t)d * NCLASS + c], xw[(size_t)s * NCLASS + c] * norm);
}

// ---------------- launch ----------------

extern "C" void kernel_launch(void* const* d_in, const int* in_sizes, int n_in,
                              void* d_out, int out_size, void* d_ws, size_t ws_size,
                              hipStream_t stream) {
    const float* x    = (const float*)d_in[0];
    const float* W    = (const float*)d_in[1];
    const float* bias = (const float*)d_in[2];
    const int*   ei   = (const int*)d_in[3];   // edge_index, shape (2, E) row-major
    const int*   src  = ei;
    const int*   dst  = ei + N_EDGES;
    float* out = (float*)d_out;

    // workspace layout: dinv (padded to 102400 floats) | xw (100000*64 floats)
    float* dinv = (float*)d_ws;
    float* xw   = dinv + 102400;

    k_init_deg<<<(N_NODES + 255) / 256, 256, 0, stream>>>(dinv);
    k_count  <<<(N_EDGES + 255) / 256, 256, 0, stream>>>(dst, dinv);
    k_dinv   <<<(N_NODES + 255) / 256, 256, 0, stream>>>(dinv);

    k_gemm_wmma<<<(MTILES + WAVES_PER_BLOCK - 1) / WAVES_PER_BLOCK, 256, 0, stream>>>(x, W, xw);

    k_out_init<<<(N_NODES * NCLASS + 255) / 256, 256, 0, stream>>>(xw, dinv, bias, out);

    long long tot = (long long)N_EDGES * NCLASS;
    k_scatter<<<(unsigned)((tot + 255) / 256), 256, 0, stream>>>(src, dst, xw, dinv, out);
}